// MyModel_61933428415605
// MI455X (gfx1250) — compile-verified
//
#include <hip/hip_runtime.h>
#include <hip/hip_bf16.h>
#include <stdint.h>

typedef __attribute__((ext_vector_type(2))) float v2f;
typedef __attribute__((ext_vector_type(8))) float v8f;

// ---- CDNA5 async copy: HBM -> LDS, 512 floats (2 KB) per chunk, padded layout.
// 64-byte data chunk c lands at LDS offset 80*c (16B pad per 64B) so that the
// A-fragment ds_load_b64 pattern (word index 20m + k) is bank-conflict free.
__device__ __forceinline__ void async_copy_chunk(const float* gsrc, uint32_t lds_off, int lane) {
#pragma unroll
  for (int j = 0; j < 4; ++j) {
    int idx = lane + 32 * j;                                  // 16-byte unit, 0..127
    uint32_t laddr = lds_off + 80u * (uint32_t)(idx >> 2) + 16u * (uint32_t)(idx & 3);
    unsigned long long gaddr = (unsigned long long)(const char*)gsrc + 16ull * (unsigned)idx;
    asm volatile("global_load_async_to_lds_b128 %0, %1, off"
                 :: "v"(laddr), "v"(gaddr) : "memory");
  }
}
__device__ __forceinline__ void wait_async_le4() { asm volatile("s_wait_asynccnt 4" ::: "memory"); }
__device__ __forceinline__ void wait_async_0()  { asm volatile("s_wait_asynccnt 0" ::: "memory"); }

__device__ __forceinline__ float rdlane(float v, int l) {
  return __builtin_bit_cast(float, __builtin_amdgcn_readlane(__builtin_bit_cast(int, v), l));
}

// Scan one 16x16 (row-major, flat=16m+n) fp32 tile with 8 chained
// v_wmma_f32_16x16x4_f32 ops; S = running prefix entering the tile (uniform).
// Returns running prefix after the tile (Z[15][15]).
__device__ __forceinline__ float scan_tile(const unsigned char* tbase, float* otile,
                                           float S, const v2f U[4], const v2f Lt[4],
                                           int lane) {
  const int m  = lane & 15;
  const int hi = lane >> 4;          // 0: K/M-even half, 1: odd half (A/B layout)
  // A fragments of X from padded LDS: lane holds X[m][4kb + 2*hi + {0,1}]
  v2f a[4];
#pragma unroll
  for (int kb = 0; kb < 4; ++kb)
    a[kb] = *reinterpret_cast<const v2f*>(tbase + 80 * m + 4 * (4 * kb + 2 * hi));

  // Chain 1: Y = X @ U + S   (per-row inclusive scan, carry folded into C)
  v8f acc;
#pragma unroll
  for (int j = 0; j < 8; ++j) acc[j] = S;
#pragma unroll
  for (int kb = 0; kb < 4; ++kb)
    acc = __builtin_amdgcn_wmma_f32_16x16x4_f32(false, a[kb], false, U[kb],
                                                (short)0, acc, false, false);

  // Row totals r[k] = Y[k][15] - S  (column 15 lives in lanes 15 / 31)
  float r[16];
#pragma unroll
  for (int j = 0; j < 8; ++j) {
    r[j]     = rdlane(acc[j], 15) - S;
    r[j + 8] = rdlane(acc[j], 31) - S;
  }

  // Chain 2: Z = Lstrict @ (r broadcast over N) + Y  (adds preceding-row totals)
#pragma unroll
  for (int kb = 0; kb < 4; ++kb) {
    v2f bs;
    bs[0] = hi ? r[4 * kb + 2] : r[4 * kb + 0];
    bs[1] = hi ? r[4 * kb + 3] : r[4 * kb + 1];
    acc = __builtin_amdgcn_wmma_f32_16x16x4_f32(false, Lt[kb], false, bs,
                                                (short)0, acc, false, false);
  }

  // Store D layout: vgpr j -> row (j + 8*hi), col (lane&15); two 64B segments/store
#pragma unroll
  for (int j = 0; j < 8; ++j)
    otile[16 * (j + 8 * hi) + (lane & 15)] = acc[j];

  return rdlane(acc[7], 31);  // Z[15][15] = prefix after this tile
}

__global__ __launch_bounds__(256) void cumsum_wmma_kernel(const float* __restrict__ x,
                                                          float* __restrict__ out,
                                                          int n_rows, int n_cols) {
  // 8 waves/block, one row per wave; per-wave double buffer: 2 x 2560 B (padded 2 KB)
  __shared__ alignas(16) unsigned char smem[8 * 2 * 2560];
  const int lane = threadIdx.x & 31;
  const int wave = threadIdx.x >> 5;
  const int row  = blockIdx.x * 8 + wave;
  if (row >= n_rows) return;  // whole-wave uniform exit; EXEC stays all-ones

  const int hi = lane >> 4;
  const int m  = lane & 15;
  // Constant fragments (A/B layout: element j of pair holds K = 4kb + j + 2*hi):
  //   U  (B operand): U[k][n] = (k <= n)  -> inclusive row scan
  //   Lt (A operand): L[m][k] = (k <  m)  -> strict lower triangular
  v2f U[4], Lt[4];
#pragma unroll
  for (int kb = 0; kb < 4; ++kb)
#pragma unroll
    for (int j = 0; j < 2; ++j) {
      int k = 4 * kb + j + 2 * hi;
      U[kb][j]  = (k <= m) ? 1.0f : 0.0f;   // n == lane&15 == m
      Lt[kb][j] = (k <  m) ? 1.0f : 0.0f;
    }

  const float* grow = x + (size_t)row * (size_t)n_cols;
  float* orow = out + (size_t)row * (size_t)n_cols;
  unsigned char* wbase = smem + wave * 5120;
  // Flat LDS addresses carry the byte offset in bits [31:0] (SHARED_BASE in [63:32])
  const uint32_t woff = (uint32_t)(uintptr_t)wbase;

  const int nchunks = n_cols / 512;  // 32 chunks of 512 floats
  async_copy_chunk(grow, woff, lane);

  float S = 0.0f;
  for (int c = 0; c < nchunks; ++c) {
    const int phase = c & 1;
    if (c + 1 < nchunks) {
      async_copy_chunk(grow + (size_t)(c + 1) * 512, woff + (uint32_t)(((c + 1) & 1) * 2560), lane);
      wait_async_le4();   // chunk c's 4 copies done; chunk c+1's 4 still in flight
    } else {
      wait_async_0();
    }
    const unsigned char* buf = wbase + phase * 2560;
#pragma unroll
    for (int t = 0; t < 2; ++t)
      S = scan_tile(buf + t * 1280, orow + (size_t)c * 512 + t * 256, S, U, Lt, lane);
  }
}

extern "C" void kernel_launch(void* const* d_in, const int* in_sizes, int n_in,
                              void* d_out, int out_size, void* d_ws, size_t ws_size,
                              hipStream_t stream) {
  (void)n_in; (void)d_ws; (void)ws_size; (void)out_size;
  const float* x = (const float*)d_in[0];
  float* out = (float*)d_out;
  const int n_cols = 16384;
  const int n_rows = in_sizes[0] / n_cols;      // 4096
  const int blocks = (n_rows + 7) / 8;          // 512 blocks x 256 threads (8 waves)
  hipLaunchKernelGGL(cumsum_wmma_kernel, dim3(blocks), dim3(256), 0, stream,
                     x, out, n_rows, n_cols);
}